// GAT_28784870818245
// MI455X (gfx1250) — compile-verified
//
#include <hip/hip_runtime.h>

#define ALPHA_SLOPE 0.2f
#define NEG_BIG (-9e15f)
#define Bsz 8
#define Nn 2048
#define Ff 256

typedef float v2f __attribute__((ext_vector_type(2)));
typedef float v4f __attribute__((ext_vector_type(4)));
typedef float v8f __attribute__((ext_vector_type(8)));

// ---------------------------------------------------------------------------
// Kernel 1: Wh[b,n,f] = sum_k h[b,n,k] * W[k,f]
// One wave per 16x16 output tile, V_WMMA_F32_16X16X4_F32, K advances by 4.
// A 16x4 f32 layout: lane (0-15) holds row M=lane, v0=K(+2*hi), v1=K+1(+2*hi)
// C/D 16x16 f32 layout: VGPR r -> M = r + 8*hi, N = lane&15
// ---------------------------------------------------------------------------
__global__ __launch_bounds__(256) void gat_wh_gemm(const float* __restrict__ h,
                                                   const float* __restrict__ W,
                                                   float* __restrict__ Wh) {
  const int lane = threadIdx.x & 31;
  const int wave = threadIdx.x >> 5;
  const int tile = blockIdx.x * 8 + wave;
  const int fTiles = Ff / 16;            // 16
  const int nTiles = Nn / 16;            // 128
  const int b  = tile / (nTiles * fTiles);
  const int r  = tile % (nTiles * fTiles);
  const int n0 = (r / fTiles) * 16;
  const int f0 = (r % fTiles) * 16;

  const int hi = lane >> 4;              // 0 or 1 (half-wave)
  const int l  = lane & 15;

  const float* __restrict__ arow = h + ((size_t)b * Nn + (n0 + l)) * Ff;
  v8f acc = {};
  #pragma unroll 4
  for (int k = 0; k < Ff; k += 4) {
    const int ka = k + 2 * hi;
    v2f av = *(const v2f*)(arow + ka);                 // contiguous along K
    v2f bv;
    bv.x = W[(size_t)(ka + 0) * Ff + f0 + l];
    bv.y = W[(size_t)(ka + 1) * Ff + f0 + l];
    acc = __builtin_amdgcn_wmma_f32_16x16x4_f32(false, av, false, bv,
                                                (short)0, acc, false, false);
  }
  float* __restrict__ orow = Wh + ((size_t)b * Nn + n0) * Ff + f0;
  #pragma unroll
  for (int rr = 0; rr < 8; ++rr) {
    const int M = rr + 8 * hi;
    orow[(size_t)M * Ff + l] = acc[rr];
  }
}

// ---------------------------------------------------------------------------
// Kernel 2: Wh1 = Wh . a[:F], Wh2 = Wh . a[F:]  (one thread per (b,n) row)
// ---------------------------------------------------------------------------
__global__ __launch_bounds__(256) void gat_proj(const float* __restrict__ Wh,
                                                const float* __restrict__ a,
                                                float* __restrict__ Wh1,
                                                float* __restrict__ Wh2) {
  const int idx = blockIdx.x * blockDim.x + threadIdx.x;   // over B*N
  if (idx >= Bsz * Nn) return;
  const float* __restrict__ row = Wh + (size_t)idx * Ff;
  float s1 = 0.f, s2 = 0.f;
  #pragma unroll 4
  for (int f = 0; f < Ff; f += 4) {
    v4f w  = *(const v4f*)(row + f);
    v4f a1 = *(const v4f*)(a + f);
    v4f a2 = *(const v4f*)(a + Ff + f);
    s1 += w.x * a1.x + w.y * a1.y + w.z * a1.z + w.w * a1.w;
    s2 += w.x * a2.x + w.y * a2.y + w.z * a2.z + w.w * a2.w;
  }
  Wh1[idx] = s1;
  Wh2[idx] = s2;
}

// ---------------------------------------------------------------------------
// Kernel 3: masked row softmax.
// attention[b,n,m] = softmax_m( adj>0 ? leaky_relu(Wh1[b,n]+Wh2[b,m]) : -9e15 )
// One 256-thread block per row; 8 scores/thread held in registers.
// ---------------------------------------------------------------------------
__global__ __launch_bounds__(256) void gat_softmax(const float* __restrict__ Wh1,
                                                   const float* __restrict__ Wh2,
                                                   const int* __restrict__ adj,
                                                   float* __restrict__ attn) {
  const int row = blockIdx.x;            // over B*N
  const int b   = row / Nn;
  const int tid = threadIdx.x;

  const float w1 = Wh1[row];
  const float* __restrict__ w2   = Wh2 + (size_t)b * Nn;
  const int*   __restrict__ arow = adj + (size_t)row * Nn;

  float s[8];
  float lmax = NEG_BIG;
  #pragma unroll
  for (int i = 0; i < 8; ++i) {
    const int m = tid + i * 256;
    float e = w1 + w2[m];
    e = (e > 0.f) ? e : ALPHA_SLOPE * e;
    const float sc = (arow[m] > 0) ? e : NEG_BIG;
    s[i] = sc;
    lmax = fmaxf(lmax, sc);
  }

  __shared__ float red[256];
  red[tid] = lmax;
  __syncthreads();
  for (int st = 128; st > 0; st >>= 1) {
    if (tid < st) red[tid] = fmaxf(red[tid], red[tid + st]);
    __syncthreads();
  }
  const float rmax = red[0];
  __syncthreads();

  float lsum = 0.f;
  #pragma unroll
  for (int i = 0; i < 8; ++i) {
    s[i] = __expf(s[i] - rmax);          // masked entries underflow to 0
    lsum += s[i];
  }
  red[tid] = lsum;
  __syncthreads();
  for (int st = 128; st > 0; st >>= 1) {
    if (tid < st) red[tid] += red[tid + st];
    __syncthreads();
  }
  const float inv = 1.0f / red[0];

  float* __restrict__ orow = attn + (size_t)row * Nn;
  #pragma unroll
  for (int i = 0; i < 8; ++i) orow[tid + i * 256] = s[i] * inv;
}

// ---------------------------------------------------------------------------
// Kernel 4: h_prime = attention @ Wh ; out = h + elu(h_prime)
// One wave per 16x16 tile, K over 2048 in steps of 4 (f32 WMMA).
// attention rows are served from L2 (each batch's 16 MB matrix fits easily).
// ---------------------------------------------------------------------------
__global__ __launch_bounds__(256) void gat_out_gemm(const float* __restrict__ attn,
                                                    const float* __restrict__ Wh,
                                                    const float* __restrict__ h,
                                                    float* __restrict__ out) {
  const int lane = threadIdx.x & 31;
  const int wave = threadIdx.x >> 5;
  const int tile = blockIdx.x * 8 + wave;
  const int fTiles = Ff / 16;
  const int nTiles = Nn / 16;
  const int b  = tile / (nTiles * fTiles);
  const int r  = tile % (nTiles * fTiles);
  const int n0 = (r / fTiles) * 16;
  const int f0 = (r % fTiles) * 16;

  const int hi = lane >> 4;
  const int l  = lane & 15;

  const float* __restrict__ arow = attn + ((size_t)b * Nn + (n0 + l)) * Nn;
  const float* __restrict__ bcol = Wh + (size_t)b * Nn * Ff + f0 + l;

  v8f acc = {};
  #pragma unroll 4
  for (int k = 0; k < Nn; k += 4) {
    const int ka = k + 2 * hi;
    v2f av = *(const v2f*)(arow + ka);
    v2f bv;
    bv.x = bcol[(size_t)(ka + 0) * Ff];
    bv.y = bcol[(size_t)(ka + 1) * Ff];
    acc = __builtin_amdgcn_wmma_f32_16x16x4_f32(false, av, false, bv,
                                                (short)0, acc, false, false);
  }

  #pragma unroll
  for (int rr = 0; rr < 8; ++rr) {
    const int M = rr + 8 * hi;
    const size_t idx = ((size_t)b * Nn + n0 + M) * Ff + f0 + l;
    const float hp = acc[rr];
    const float e = (hp > 0.f) ? hp : (__expf(hp) - 1.0f);   // elu
    out[idx] = h[idx] + e;
  }
}

// ---------------------------------------------------------------------------
extern "C" void kernel_launch(void* const* d_in, const int* in_sizes, int n_in,
                              void* d_out, int out_size, void* d_ws, size_t ws_size,
                              hipStream_t stream) {
  const float* h   = (const float*)d_in[0];   // [B,N,F] f32
  const int*   adj = (const int*)d_in[1];     // [B,N,N] i32
  const float* W   = (const float*)d_in[2];   // [F,F]  f32
  const float* a   = (const float*)d_in[3];   // [2F,1] f32

  float* out  = (float*)d_out;                          // [B,N,F]
  float* attn = (float*)d_out + (size_t)Bsz * Nn * Ff;  // [B,N,N]

  float* Wh  = (float*)d_ws;                            // 16 MB
  float* Wh1 = Wh + (size_t)Bsz * Nn * Ff;              // 64 KB
  float* Wh2 = Wh1 + (size_t)Bsz * Nn;                  // 64 KB

  const int tiles = Bsz * (Nn / 16) * (Ff / 16);        // 16384, 8 waves/block

  gat_wh_gemm<<<tiles / 8, 256, 0, stream>>>(h, W, Wh);
  gat_proj<<<(Bsz * Nn + 255) / 256, 256, 0, stream>>>(Wh, a, Wh1, Wh2);
  gat_softmax<<<Bsz * Nn, 256, 0, stream>>>(Wh1, Wh2, adj, attn);
  gat_out_gemm<<<tiles / 8, 256, 0, stream>>>(attn, Wh, h, out);
}